// YoloLoss_47296179863926
// MI455X (gfx1250) — compile-verified
//
#include <hip/hip_runtime.h>
#include <math.h>

typedef __attribute__((ext_vector_type(2))) float v2f;
typedef __attribute__((ext_vector_type(8))) float v8f;

#define SDIM   7
#define NBOX   2
#define NCLS   20
#define PRED_C (NBOX * 5 + NCLS)  // 30
#define TGT_C  (5 + NCLS)         // 25
#define L_OBJ_W   5.0f
#define L_NOOBJ_W 0.5f
#define EPS_IOU 1e-6f
#define EPS_WH  1e-6f
#define NBLOCKS_MAIN 1024

__device__ __forceinline__ float sigmoidf_(float x) {
    return 1.0f / (1.0f + expf(-x));
}
__device__ __forceinline__ float clip01(float x) {
    return fminf(fmaxf(x, 0.0f), 1.0f);
}

// Full-precision f32 reduction of all 32 lane values of a wave using
// V_WMMA_F32_16X16X4_F32 with an all-ones B matrix.
//   Step 1: A[m][0]=x(lane m), A[m][2]=x(lane m+16), rest 0.
//           D[m][n] = x[m] + x[m+16]  (row sums broadcast over n).
//           Per-lane sum of the 8 D VGPRs: lanes 0-15 -> S_low (sum of rows 0-7),
//           lanes 16-31 -> S_high (rows 8-15).
//   Step 2: place S_low on lane 0 (A[0][0]) and S_high on lane 16 (A[0][2]);
//           D[0][n] = S_low + S_high = wave total, visible in c[0] on lanes 0-15.
__device__ __forceinline__ float wave_sum_wmma(float x, int lane) {
    v2f ones; ones.x = 1.0f; ones.y = 1.0f;
    v2f a;    a.x = x;       a.y = 0.0f;
    v8f c = {};
    c = __builtin_amdgcn_wmma_f32_16x16x4_f32(false, a, false, ones,
                                              (short)0, c, false, false);
    float s = 0.0f;
#pragma unroll
    for (int r = 0; r < 8; ++r) s += c[r];
    v2f a2; a2.x = ((lane & 15) == 0) ? s : 0.0f; a2.y = 0.0f;
    v8f c2 = {};
    c2 = __builtin_amdgcn_wmma_f32_16x16x4_f32(false, a2, false, ones,
                                               (short)0, c2, false, false);
    return c2[0];  // wave total, valid on lanes 0..15
}

__global__ void __launch_bounds__(256)
yolo_loss_main(const float* __restrict__ pred,
               const float* __restrict__ target,
               float* __restrict__ partials,
               int ncells) {
    float acc = 0.0f;
    const int stride = gridDim.x * blockDim.x;
    for (int cell = blockIdx.x * blockDim.x + threadIdx.x; cell < ncells;
         cell += stride) {
        // ---- load pred cell (30 f32, 8-byte aligned -> b64 loads) ----
        const float2* pp =
            reinterpret_cast<const float2*>(pred + (size_t)cell * PRED_C);
        float p[PRED_C];
#pragma unroll
        for (int k = 0; k < PRED_C / 2; ++k) {
            float2 v = pp[k];
            p[2 * k] = v.x;
            p[2 * k + 1] = v.y;
        }
        // ---- load target cell (25 f32) ----
        const float* tp = target + (size_t)cell * TGT_C;
        float t[TGT_C];
#pragma unroll
        for (int k = 0; k < TGT_C; ++k) t[k] = tp[k];

        // ---- IoU of both predicted boxes vs (clipped) target box ----
        const float tx = clip01(t[1]);
        const float ty = clip01(t[2]);
        const float tw = clip01(t[3]);
        const float th = clip01(t[4]);
        const float twh = tw * th;

        float iou[NBOX], sx[NBOX], sy[NBOX], s0[NBOX];
#pragma unroll
        for (int b = 0; b < NBOX; ++b) {
            const int o = b * 5;
            s0[b] = sigmoidf_(p[o + 0]);
            const float px = sigmoidf_(p[o + 1]);
            const float py = sigmoidf_(p[o + 2]);
            sx[b] = px;
            sy[b] = py;
            const float pw = fabsf(p[o + 3]);
            const float ph = fabsf(p[o + 4]);
            const float ix0 = fmaxf(px - 0.5f * pw, tx - 0.5f * tw);
            const float iy0 = fmaxf(py - 0.5f * ph, ty - 0.5f * th);
            const float ix1 = fminf(px + 0.5f * pw, tx + 0.5f * tw);
            const float iy1 = fminf(py + 0.5f * ph, ty + 0.5f * th);
            const float inter =
                fmaxf(ix1 - ix0, 0.0f) * fmaxf(iy1 - iy0, 0.0f);
            const float uni = pw * ph + twh - inter + EPS_IOU;
            iou[b] = inter / uni;
        }

        // argmax with first-max tie-break (ties pick box 0)
        const int bi = (iou[1] > iou[0]) ? 1 : 0;
        const float bsx = bi ? sx[1] : sx[0];
        const float bsy = bi ? sy[1] : sy[0];
        const float bs0 = bi ? s0[1] : s0[0];
        const float biou = bi ? iou[1] : iou[0];
        const float bw = bi ? p[8] : p[3];
        const float bh = bi ? p[9] : p[4];

        // xy loss vs RAW target coords (no clip outside IoU)
        const float dx = bsx - t[1];
        const float dy = bsy - t[2];
        const float xy_loss = dx * dx + dy * dy;

        const float dw = sqrtf(fabsf(bw + EPS_WH)) - sqrtf(fabsf(t[3] + EPS_WH));
        const float dh = sqrtf(fabsf(bh + EPS_WH)) - sqrtf(fabsf(t[4] + EPS_WH));
        const float wh_loss = dw * dw + dh * dh;

        const float dconf = bs0 - biou;
        const float conf_obj = dconf * dconf;

        // stable softmax over 20 class logits
        float m = p[NBOX * 5];
#pragma unroll
        for (int k = 1; k < NCLS; ++k) m = fmaxf(m, p[NBOX * 5 + k]);
        float e[NCLS];
        float esum = 0.0f;
#pragma unroll
        for (int k = 0; k < NCLS; ++k) {
            e[k] = expf(p[NBOX * 5 + k] - m);
            esum += e[k];
        }
        const float inv = 1.0f / esum;
        float class_loss = 0.0f;
#pragma unroll
        for (int k = 0; k < NCLS; ++k) {
            const float d = e[k] * inv - t[5 + k];
            class_loss += d * d;
        }

        const float obj_term =
            L_OBJ_W * (xy_loss + wh_loss) + conf_obj + class_loss;
        const float noobj_term =
            L_NOOBJ_W * (s0[0] * s0[0] + s0[1] * s0[1]);

        acc += (t[0] == 1.0f) ? obj_term : noobj_term;
    }

    // ---- deterministic block reduction: WMMA per wave, then LDS ----
    const int lane = threadIdx.x & 31;
    const int wave = threadIdx.x >> 5;
    const float wtot = wave_sum_wmma(acc, lane);

    __shared__ float smem[8];  // 256 threads -> 8 waves
    if (lane == 0) smem[wave] = wtot;
    __syncthreads();
    if (threadIdx.x == 0) {
        float b = 0.0f;
#pragma unroll
        for (int w = 0; w < 8; ++w) b += smem[w];
        partials[blockIdx.x] = b;
    }
}

__global__ void __launch_bounds__(32)
yolo_loss_finish(const float* __restrict__ partials,
                 float* __restrict__ out, int nparts, float invN) {
    const int lane = threadIdx.x;  // exactly one wave
    float s = 0.0f;
    for (int i = lane; i < nparts; i += 32) s += partials[i];
    const float tot = wave_sum_wmma(s, lane);
    if (lane == 0) out[0] = tot * invN;
}

extern "C" void kernel_launch(void* const* d_in, const int* in_sizes, int n_in,
                              void* d_out, int out_size, void* d_ws,
                              size_t ws_size, hipStream_t stream) {
    const float* pred = (const float*)d_in[0];
    const float* target = (const float*)d_in[1];
    float* out = (float*)d_out;
    float* partials = (float*)d_ws;

    const int n = in_sizes[0] / (SDIM * SDIM * PRED_C);  // 16384
    const int ncells = n * SDIM * SDIM;                  // 802816

    int nblocks = NBLOCKS_MAIN;
    if (ws_size < (size_t)nblocks * sizeof(float)) {
        nblocks = (int)(ws_size / sizeof(float));
        if (nblocks < 1) nblocks = 1;
    }

    yolo_loss_main<<<nblocks, 256, 0, stream>>>(pred, target, partials, ncells);
    yolo_loss_finish<<<1, 32, 0, stream>>>(partials, out, nblocks,
                                           1.0f / (float)n);
}